// GCCF_35158602285525
// MI455X (gfx1250) — compile-verified
//
#include <hip/hip_runtime.h>
#include <cstdint>
#include <cstddef>

#define EMB 64

// ---------------------------------------------------------------------------
// K0: zero workspace (accumulators + flags). Must run every call: the harness
// graph-replays kernel_launch and the previous replay left sums behind.
// ---------------------------------------------------------------------------
__global__ void k_zero(uint4* __restrict__ p, long n16) {
  long i = (long)blockIdx.x * blockDim.x + threadIdx.x;
  long stride = (long)gridDim.x * blockDim.x;
  uint4 z; z.x = 0u; z.y = 0u; z.z = 0u; z.w = 0u;
  for (; i < n16; i += stride) p[i] = z;
}

// ---------------------------------------------------------------------------
// K1: mark which user rows / item rows are actually needed by the batch.
// Benign races: everyone stores the byte 1.
// ---------------------------------------------------------------------------
__global__ void k_flags(const int* __restrict__ user,
                        const int* __restrict__ item_i,
                        const int* __restrict__ item_j,
                        unsigned char* __restrict__ flag_u,
                        unsigned char* __restrict__ flag_i, int B) {
  int t = blockIdx.x * blockDim.x + threadIdx.x;
  if (t < B)            flag_u[user[t]]          = 1;
  else if (t < 2 * B)   flag_i[item_i[t - B]]    = 1;
  else if (t < 3 * B)   flag_i[item_j[t - 2*B]]  = 1;
}

// ---------------------------------------------------------------------------
// K2: edge scatter. 16 lanes cooperate on one edge (lane -> one float4 of the
// 64-float row). Both SpMM directions handled in one pass over the edge list.
// Gathered rows are staged LDS-side with the CDNA5 async copy engine
// (global_load_async_to_lds_b128, tracked by ASYNCcnt), then scattered with
// f32 atomic adds that resolve in the 192MB L2 (accumulators are L2-resident).
// ---------------------------------------------------------------------------
__global__ __launch_bounds__(256)
void k_edges(const float* __restrict__ embed_user,
             const float* __restrict__ embed_item,
             const float* __restrict__ uv_vals,
             const float* __restrict__ vu_vals,
             const int*   __restrict__ uv_rows,
             const int*   __restrict__ uv_cols,
             const unsigned char* __restrict__ flag_u,
             const unsigned char* __restrict__ flag_i,
             float* __restrict__ agg_u,
             float* __restrict__ agg_i,
             int nnz) {
  // [wave][direction][lane][float4] : per-wave private staging, 8 KB / block
  __shared__ float stage[8][2][32][4];

  int t    = blockIdx.x * 256 + threadIdx.x;
  int wave = threadIdx.x >> 5;
  int lane = threadIdx.x & 31;
  int edge = t >> 4;   // 16 lanes per edge
  int sub  = t & 15;   // which float4 of the 64-float row
  if (edge >= nnz) return;

  int r = uv_rows[edge];
  int c = uv_cols[edge];
  bool doU = flag_u[r] != 0;   // scatter target user row needed?
  bool doI = flag_i[c] != 0;   // scatter target item row needed?

  // Workgroup-relative LDS byte offsets (low 32 bits of generic shared ptr).
  uint32_t ldsU = (uint32_t)(uintptr_t)&stage[wave][0][lane][0];
  uint32_t ldsI = (uint32_t)(uintptr_t)&stage[wave][1][lane][0];

  if (doU) {
    uint32_t voff = ((uint32_t)c * EMB + (uint32_t)sub * 4u) * 4u; // bytes
    asm volatile("global_load_async_to_lds_b128 %0, %1, %2"
                 :: "v"(ldsU), "v"(voff), "s"(embed_item) : "memory");
  }
  if (doI) {
    uint32_t voff = ((uint32_t)r * EMB + (uint32_t)sub * 4u) * 4u; // bytes
    asm volatile("global_load_async_to_lds_b128 %0, %1, %2"
                 :: "v"(ldsI), "v"(voff), "s"(embed_user) : "memory");
  }
  asm volatile("s_wait_asynccnt 0" ::: "memory");

  if (doU) {
    float v = uv_vals[edge];
    const float4 x = *(const float4*)&stage[wave][0][lane][0];
    float* dst = agg_u + (size_t)r * EMB + (size_t)sub * 4;
    __hip_atomic_fetch_add(dst + 0, v * x.x, __ATOMIC_RELAXED, __HIP_MEMORY_SCOPE_AGENT);
    __hip_atomic_fetch_add(dst + 1, v * x.y, __ATOMIC_RELAXED, __HIP_MEMORY_SCOPE_AGENT);
    __hip_atomic_fetch_add(dst + 2, v * x.z, __ATOMIC_RELAXED, __HIP_MEMORY_SCOPE_AGENT);
    __hip_atomic_fetch_add(dst + 3, v * x.w, __ATOMIC_RELAXED, __HIP_MEMORY_SCOPE_AGENT);
  }
  if (doI) {
    float v = vu_vals[edge];
    const float4 x = *(const float4*)&stage[wave][1][lane][0];
    float* dst = agg_i + (size_t)c * EMB + (size_t)sub * 4;
    __hip_atomic_fetch_add(dst + 0, v * x.x, __ATOMIC_RELAXED, __HIP_MEMORY_SCOPE_AGENT);
    __hip_atomic_fetch_add(dst + 1, v * x.y, __ATOMIC_RELAXED, __HIP_MEMORY_SCOPE_AGENT);
    __hip_atomic_fetch_add(dst + 2, v * x.z, __ATOMIC_RELAXED, __HIP_MEMORY_SCOPE_AGENT);
    __hip_atomic_fetch_add(dst + 3, v * x.w, __ATOMIC_RELAXED, __HIP_MEMORY_SCOPE_AGENT);
  }
}

// ---------------------------------------------------------------------------
// K3: batch gather. 32 lanes per 128-float output row.
// out row = [ embed[idx][0:64] | agg[idx] + embed[idx]*d[idx] ]
// Rows ordered: users (B), item_i (B), item_j (B).
// ---------------------------------------------------------------------------
__global__ __launch_bounds__(256)
void k_gather(const float* __restrict__ eu, const float* __restrict__ ei,
              const float* __restrict__ di, const float* __restrict__ dj,
              const float* __restrict__ agg_u, const float* __restrict__ agg_i,
              const int* __restrict__ user, const int* __restrict__ item_i,
              const int* __restrict__ item_j,
              float* __restrict__ out, int B) {
  int t   = blockIdx.x * blockDim.x + threadIdx.x;
  int row = t >> 5;
  int ch  = t & 31;          // 32 float4 chunks per 128-float row
  if (row >= 3 * B) return;

  int which = row / B;
  int b = row - which * B;
  int idx;
  const float *emb, *agg, *dd;
  if (which == 0)      { idx = user[b];   emb = eu; agg = agg_u; dd = di; }
  else if (which == 1) { idx = item_i[b]; emb = ei; agg = agg_i; dd = dj; }
  else                 { idx = item_j[b]; emb = ei; agg = agg_i; dd = dj; }

  float* orow = out + (size_t)row * (2 * EMB);
  if (ch < 16) {
    const float4 x = *(const float4*)(emb + (size_t)idx * EMB + (size_t)ch * 4);
    *(float4*)(orow + ch * 4) = x;
  } else {
    int k = ch - 16;
    const float4 x = *(const float4*)(emb + (size_t)idx * EMB + (size_t)k * 4);
    const float4 a = *(const float4*)(agg + (size_t)idx * EMB + (size_t)k * 4);
    float d = dd[idx];
    float4 ro;
    ro.x = a.x + x.x * d;
    ro.y = a.y + x.y * d;
    ro.z = a.z + x.z * d;
    ro.w = a.w + x.w * d;
    *(float4*)(orow + EMB + k * 4) = ro;
  }
}

// ---------------------------------------------------------------------------
extern "C" void kernel_launch(void* const* d_in, const int* in_sizes, int n_in,
                              void* d_out, int out_size, void* d_ws, size_t ws_size,
                              hipStream_t stream) {
  const float* embed_user = (const float*)d_in[0];
  const float* embed_item = (const float*)d_in[1];
  const float* d_i        = (const float*)d_in[2];
  const float* d_j        = (const float*)d_in[3];
  const float* uv_vals    = (const float*)d_in[4];
  const float* vu_vals    = (const float*)d_in[5];
  const int*   uv_rows    = (const int*)d_in[6];
  const int*   uv_cols    = (const int*)d_in[7];
  const int*   user       = (const int*)d_in[8];
  const int*   item_i     = (const int*)d_in[9];
  const int*   item_j     = (const int*)d_in[10];

  const int n_user = in_sizes[2];   // d_i length
  const int n_item = in_sizes[3];   // d_j length
  const int nnz    = in_sizes[4];
  const int B      = in_sizes[8];

  // Workspace layout
  float* agg_u = (float*)d_ws;                                   // n_user*64 f32
  float* agg_i = agg_u + (size_t)n_user * EMB;                   // n_item*64 f32
  unsigned char* flag_u = (unsigned char*)(agg_i + (size_t)n_item * EMB);
  unsigned char* flag_i = flag_u + n_user;

  size_t total_bytes = (size_t)(n_user + n_item) * EMB * sizeof(float)
                     + (size_t)n_user + (size_t)n_item;
  long n16 = (long)((total_bytes + 15) / 16);

  k_zero<<<2048, 256, 0, stream>>>((uint4*)d_ws, n16);

  int fthreads = 3 * B;
  k_flags<<<(fthreads + 255) / 256, 256, 0, stream>>>(user, item_i, item_j,
                                                      flag_u, flag_i, B);

  long ethreads = (long)nnz * 16;
  k_edges<<<(unsigned)((ethreads + 255) / 256), 256, 0, stream>>>(
      embed_user, embed_item, uv_vals, vu_vals, uv_rows, uv_cols,
      flag_u, flag_i, agg_u, agg_i, nnz);

  long gthreads = (long)3 * B * 32;
  k_gather<<<(unsigned)((gthreads + 255) / 256), 256, 0, stream>>>(
      embed_user, embed_item, d_i, d_j, agg_u, agg_i,
      user, item_i, item_j, (float*)d_out, B);
}